// GraphDecoder_30932354466113
// MI455X (gfx1250) — compile-verified
//
#include <hip/hip_runtime.h>
#include <hip/hip_bf16.h>

typedef float v2f __attribute__((ext_vector_type(2)));
typedef float v8f __attribute__((ext_vector_type(8)));

#define IN_DIM  1200
#define OUT_DIM 300
#define N_TILES 19          // ceil(300/16)
#define NEG_SLOPE 0.2f

// ---------------------------------------------------------------------------
// K0: transpose W [300,1200] -> Wt [1200,300] so B-tile loads are coalesced
// ---------------------------------------------------------------------------
__global__ __launch_bounds__(256) void transpose_w(const float* __restrict__ W,
                                                   float* __restrict__ Wt) {
    int i = blockIdx.x * blockDim.x + threadIdx.x;
    if (i >= OUT_DIM * IN_DIM) return;
    int o = i / IN_DIM;
    int k = i - o * IN_DIM;
    Wt[k * OUT_DIM + o] = W[i];
}

// ---------------------------------------------------------------------------
// K1: x = z @ W^T via V_WMMA_F32_16X16X4_F32. One wave per 16x16 output tile,
// K-loop of 300 WMMAs. gridDim.y = M tile (3125), gridDim.x covers 19 N tiles
// with 4 waves per block.
// ---------------------------------------------------------------------------
__global__ __launch_bounds__(128) void gemm_wmma(const float* __restrict__ z,
                                                 const float* __restrict__ Wt,
                                                 float* __restrict__ x, int N) {
    const int wave = threadIdx.x >> 5;
    const int nt   = blockIdx.x * 4 + wave;           // column tile index
    if (nt >= N_TILES) return;                        // wave-uniform
    const int lane = threadIdx.x & 31;
    const int half = lane >> 4;                       // 0 or 1
    const int l15  = lane & 15;
    const int m0   = blockIdx.y << 4;
    const int n0   = nt << 4;

    int rA = m0 + l15;            if (rA > N - 1) rA = N - 1;        // A row
    int cB = n0 + l15;            int cBc = cB < OUT_DIM ? cB : OUT_DIM - 1;

    const float* zrow = z  + (size_t)rA * IN_DIM + half * 2;          // A: b64/lane
    const float* wcol = Wt + (size_t)half * 2 * OUT_DIM + cBc;        // B: 2x b32/lane

    v8f acc = {};
    #pragma unroll 2
    for (int k0 = 0; k0 < IN_DIM; k0 += 4) {
        v2f a = *(const v2f*)(zrow + k0);             // K = k0+2*half, +1
        v2f b;
        b.x = wcol[(size_t)k0 * OUT_DIM];             // B row k0+2*half
        b.y = wcol[(size_t)k0 * OUT_DIM + OUT_DIM];   // B row k0+2*half+1
        acc = __builtin_amdgcn_wmma_f32_16x16x4_f32(
                  false, a, false, b, (short)0, acc, false, false);
    }

    // C/D layout: VGPR v -> row m0+v (lanes 0-15) / m0+v+8 (lanes 16-31), col n0+l15
    #pragma unroll
    for (int v = 0; v < 8; ++v) {
        int row = m0 + v + (half << 3);
        int col = n0 + l15;
        if (col < OUT_DIM && row < N)
            x[(size_t)row * OUT_DIM + col] = acc[v];
    }
}

// ---------------------------------------------------------------------------
// K2: per-node attention logits a_src[n] = x[n,:]·att_src, a_dst likewise.
// One wave per node, butterfly reduction.
// ---------------------------------------------------------------------------
__global__ __launch_bounds__(256) void node_dots(const float* __restrict__ x,
                                                 const float* __restrict__ att_src,
                                                 const float* __restrict__ att_dst,
                                                 float* __restrict__ a_src,
                                                 float* __restrict__ a_dst, int N) {
    int w    = (blockIdx.x * blockDim.x + threadIdx.x) >> 5;
    int lane = threadIdx.x & 31;
    if (w >= N) return;
    const float* xr = x + (size_t)w * OUT_DIM;
    float s1 = 0.f, s2 = 0.f;
    for (int c = lane; c < OUT_DIM; c += 32) {
        float xv = xr[c];
        s1 += xv * att_src[c];
        s2 += xv * att_dst[c];
    }
    for (int m = 16; m; m >>= 1) { s1 += __shfl_xor(s1, m); s2 += __shfl_xor(s2, m); }
    if (lane == 0) { a_src[w] = s1; a_dst[w] = s2; }
}

// ---------------------------------------------------------------------------
// CSR-by-destination build (avoids 255M float atomics in the aggregation):
// counts -> block scan -> top scan -> add offsets -> fill
// ---------------------------------------------------------------------------
__global__ __launch_bounds__(256) void csr_init(int* __restrict__ row_start, int N) {
    int i = blockIdx.x * blockDim.x + threadIdx.x;
    if (i <= N) row_start[i] = 1;                     // self-loop pre-count
}

__global__ __launch_bounds__(256) void csr_count(const int* __restrict__ ei, int E,
                                                 int* __restrict__ row_start) {
    int i = blockIdx.x * blockDim.x + threadIdx.x;
    if (i >= E) return;
    atomicAdd(&row_start[ei[E + i]], 1);              // dst = ei[1][i]
}

__global__ __launch_bounds__(256) void scan_block(int* __restrict__ data,
                                                  int* __restrict__ blockSums, int N) {
    __shared__ int sh[256];
    int t = threadIdx.x;
    int i = blockIdx.x * 256 + t;
    int v = (i < N) ? data[i] : 0;
    sh[t] = v;
    __syncthreads();
    for (int off = 1; off < 256; off <<= 1) {
        int add = (t >= off) ? sh[t - off] : 0;
        __syncthreads();
        sh[t] += add;
        __syncthreads();
    }
    if (i < N) data[i] = sh[t] - v;                   // exclusive
    if (t == 255) blockSums[blockIdx.x] = sh[255];
}

__global__ void scan_tops(int* __restrict__ blockSums, int nb) {
    if (blockIdx.x == 0 && threadIdx.x == 0) {
        int run = 0;
        for (int b = 0; b < nb; ++b) { int v = blockSums[b]; blockSums[b] = run; run += v; }
    }
}

__global__ __launch_bounds__(256) void scan_add(int* __restrict__ row_start,
                                                int* __restrict__ cursor,
                                                const int* __restrict__ blockSums,
                                                int N, int total) {
    int i = blockIdx.x * blockDim.x + threadIdx.x;
    if (i < N) {
        int v = row_start[i] + blockSums[i >> 8];
        row_start[i] = v;
        cursor[i]    = v;
    }
    if (i == N) row_start[N] = total;
}

__global__ __launch_bounds__(256) void csr_fill(const int* __restrict__ ei, int E, int N,
                                                int* __restrict__ cursor,
                                                int* __restrict__ csr_src) {
    int i = blockIdx.x * blockDim.x + threadIdx.x;
    int total = E + N;
    if (i >= total) return;
    int s, d;
    if (i < E) { s = ei[i]; d = ei[E + i]; }
    else       { s = d = i - E; }                      // self-loop
    int pos = atomicAdd(&cursor[d], 1);
    csr_src[pos] = s;
}

// ---------------------------------------------------------------------------
// K6: fused edge softmax + aggregation, one wave per destination node.
// Atomic-free: two strided passes (max, sum-exp) + register accumulation.
// ---------------------------------------------------------------------------
__global__ __launch_bounds__(256) void gat_aggregate(const float* __restrict__ x,
                                                     const float* __restrict__ a_src,
                                                     const float* __restrict__ a_dst,
                                                     const int* __restrict__ row_start,
                                                     const int* __restrict__ csr_src,
                                                     const float* __restrict__ bias,
                                                     float* __restrict__ out, int N) {
    int w    = (blockIdx.x * blockDim.x + threadIdx.x) >> 5;
    int lane = threadIdx.x & 31;
    if (w >= N) return;
    int   rs = row_start[w];
    int   re = row_start[w + 1];
    float ad = a_dst[w];

    float mx = -3.402823466e38f;
    for (int j = rs + lane; j < re; j += 32) {
        float e = a_src[csr_src[j]] + ad;
        e = e > 0.f ? e : NEG_SLOPE * e;
        mx = fmaxf(mx, e);
    }
    for (int m = 16; m; m >>= 1) mx = fmaxf(mx, __shfl_xor(mx, m));

    float sum = 0.f;
    for (int j = rs + lane; j < re; j += 32) {
        float e = a_src[csr_src[j]] + ad;
        e = e > 0.f ? e : NEG_SLOPE * e;
        sum += __expf(e - mx);
    }
    for (int m = 16; m; m >>= 1) sum += __shfl_xor(sum, m);
    float inv = 1.f / sum;                             // self-loop guarantees sum > 0

    float acc[10];
    #pragma unroll
    for (int t = 0; t < 10; ++t) acc[t] = 0.f;

    for (int j = rs; j < re; ++j) {
        int   s = csr_src[j];                          // uniform across wave
        float e = a_src[s] + ad;
        e = e > 0.f ? e : NEG_SLOPE * e;
        float alpha = __expf(e - mx) * inv;
        const float* xr = x + (size_t)s * OUT_DIM;
        #pragma unroll
        for (int t = 0; t < 10; ++t) {
            int c = lane + t * 32;
            if (c < OUT_DIM) acc[t] += alpha * xr[c];  // coalesced row read
        }
    }

    float* orow = out + (size_t)w * OUT_DIM;
    #pragma unroll
    for (int t = 0; t < 10; ++t) {
        int c = lane + t * 32;
        if (c < OUT_DIM) orow[c] = acc[t] + bias[c];
    }
}

// ---------------------------------------------------------------------------
// Host-side launch
// ---------------------------------------------------------------------------
extern "C" void kernel_launch(void* const* d_in, const int* in_sizes, int n_in,
                              void* d_out, int out_size, void* d_ws, size_t ws_size,
                              hipStream_t stream) {
    const float* z       = (const float*)d_in[0];
    const int*   ei      = (const int*)  d_in[1];
    const float* W       = (const float*)d_in[2];
    const float* att_src = (const float*)d_in[3];
    const float* att_dst = (const float*)d_in[4];
    const float* bias    = (const float*)d_in[5];
    float*       out     = (float*)d_out;

    const int N = in_sizes[0] / IN_DIM;      // 50000
    const int E = in_sizes[1] / 2;           // 800000
    const int total = E + N;

    // workspace layout (256B aligned slabs)
    char*  base = (char*)d_ws;
    size_t off  = 0;
    auto carve = [&](size_t bytes) -> void* {
        void* p = base + off;
        off = (off + bytes + 255) & ~(size_t)255;
        return p;
    };
    float* x         = (float*)carve((size_t)N * OUT_DIM * sizeof(float));
    float* Wt        = (float*)carve((size_t)IN_DIM * OUT_DIM * sizeof(float));
    float* a_src     = (float*)carve((size_t)N * sizeof(float));
    float* a_dst     = (float*)carve((size_t)N * sizeof(float));
    int*   row_start = (int*)  carve((size_t)(N + 1) * sizeof(int));
    int    nb        = (N + 255) / 256;
    int*   blockSums = (int*)  carve((size_t)nb * sizeof(int));
    int*   cursor    = (int*)  carve((size_t)N * sizeof(int));
    int*   csr_src   = (int*)  carve((size_t)total * sizeof(int));

    // K0: W transpose
    transpose_w<<<(OUT_DIM * IN_DIM + 255) / 256, 256, 0, stream>>>(W, Wt);

    // K1: projection GEMM on WMMA pipe
    dim3 ggrid((N_TILES + 3) / 4, (N + 15) / 16);
    gemm_wmma<<<ggrid, 128, 0, stream>>>(z, Wt, x, N);

    // K2: attention logits
    node_dots<<<(N + 7) / 8, 256, 0, stream>>>(x, att_src, att_dst, a_src, a_dst, N);

    // CSR build
    csr_init <<<(N + 1 + 255) / 256, 256, 0, stream>>>(row_start, N);
    csr_count<<<(E + 255) / 256,     256, 0, stream>>>(ei, E, row_start);
    scan_block<<<nb, 256, 0, stream>>>(row_start, blockSums, N);
    scan_tops<<<1, 32, 0, stream>>>(blockSums, nb);
    scan_add <<<(N + 1 + 255) / 256, 256, 0, stream>>>(row_start, cursor, blockSums, N, total);
    csr_fill <<<(total + 255) / 256, 256, 0, stream>>>(ei, E, N, cursor, csr_src);

    // K6: fused softmax + aggregation + bias
    gat_aggregate<<<(N + 7) / 8, 256, 0, stream>>>(x, a_src, a_dst, row_start, csr_src,
                                                   bias, out, N);
}